// BbVertLoss_76828374991464
// MI455X (gfx1250) — compile-verified
//
#include <hip/hip_runtime.h>
#include <hip/hip_bf16.h>

// Problem constants (match reference: B=16, N=40960, H=24)
#define B_   16
#define N_   40960
#define H_   24
#define NBLK 40              // blocks along N per batch
#define PTS_PER_BLOCK 1024   // NBLK * PTS_PER_BLOCK == N_
#define STEPS 32             // per wave: 32 steps * 4 points = 128 pts; 8 waves -> 1024

typedef float v2f __attribute__((ext_vector_type(2)));
typedef float v8f __attribute__((ext_vector_type(8)));

struct BoxParams {
  float pn0, pn1, pn2, px0, px1, px2;  // pred bbox min / max
  float gn0, gn1, gn2, gx0, gx1, gx2;  // gt   bbox min / max
};

// D = A(16x4) * ones(4x16) + C : accumulates per-row (per-h) sums in exact f32.
__device__ __forceinline__ v8f wmma_rowsum(v2f a, v8f c) {
  v2f ones; ones.x = 1.0f; ones.y = 1.0f;   // all-ones B: layout-invariant
  return __builtin_amdgcn_wmma_f32_16x16x4_f32(false, a, false, ones,
                                               (short)0, c, false, false);
}

// sigmoid(clamp(100*prod, -20, 20)) with the exp2 conversion folded into the
// scale constant:  exp(-t) = 2^(-t*log2e), bounds scale to +/-20*log2e.
// mul + med3 + v_exp_f32 + add + v_rcp_f32 : 5 ops, no guards.
__device__ __forceinline__ float sigmoid_clamped(float prod) {
  float w = prod * -144.26950408889634f;                    // -100*log2(e)
  w = fminf(28.853900817779268f, fmaxf(-28.853900817779268f, w));
  return __builtin_amdgcn_rcpf(1.0f + __builtin_amdgcn_exp2f(w));
}

// Natural log for args in [1e-8, ~2]: never denormal, never huge, so skip the
// libm denorm/overflow guards: raw v_log_f32 (log2) * ln2.
__device__ __forceinline__ float fast_ln(float x) {
  return __builtin_amdgcn_logf(x) * 0.6931471805599453f;
}

__device__ __forceinline__ void eval_point(float x0, float x1, float x2,
                                           const BoxParams& bp, float mask,
                                           float& o_pred, float& o_gt,
                                           float& o_tp, float& o_ce) {
  // soft inside-pred score
  float q0 = (bp.pn0 - x0) * (x0 - bp.px0);
  float q1 = (bp.pn1 - x1) * (x1 - bp.px1);
  float q2 = (bp.pn2 - x2) * (x2 - bp.px2);
  float s0 = sigmoid_clamped(q0);
  float s1 = sigmoid_clamped(q1);
  float s2 = sigmoid_clamped(q2);
  float pred = fminf(s0, fminf(s1, s2));                    // -> v_min3_num_f32
  // hard inside-gt indicator
  float g0 = (bp.gn0 - x0) * (x0 - bp.gx0);
  float g1 = (bp.gn1 - x1) * (x1 - bp.gx1);
  float g2 = (bp.gn2 - x2) * (x2 - bp.gx2);
  bool  in = (g0 > 0.0f) && (g1 > 0.0f) && (g2 > 0.0f);
  float gt = in ? 1.0f : 0.0f;
  // cross entropy: gt is exactly 0 or 1, so select the branch, one log total.
  float carg = in ? (pred + 1e-8f) : (1.0f - pred + 1e-8f);
  float ce = -fast_ln(carg);
  o_pred = pred * mask;
  o_gt   = gt * mask;
  o_tp   = pred * gt * mask;
  o_ce   = ce * mask;
}

__device__ __forceinline__ BoxParams load_box(const float* __restrict__ Pred,
                                              const float* __restrict__ Gt,
                                              int b, int h) {
  const float* p = Pred + (size_t)(b * H_ + h) * 6;  // [min.xyz, max.xyz]
  const float* g = Gt   + (size_t)(b * H_ + h) * 6;
  BoxParams bp;
  bp.pn0 = p[0]; bp.pn1 = p[1]; bp.pn2 = p[2];
  bp.px0 = p[3]; bp.px1 = p[4]; bp.px2 = p[5];
  bp.gn0 = g[0]; bp.gn1 = g[1]; bp.gn2 = g[2];
  bp.gx0 = g[3]; bp.gx1 = g[4]; bp.gx2 = g[5];
  return bp;
}

// Stage 1: fused point pass. WMMA rows = box index h, K = 4 points per step.
// Lanes 0-15: h = lane, points (n, n+1) -> A slots K0,K1.
// Lanes 16-31: h = lane-16, points (n+2, n+3) -> A slots K2,K3.
// Group0 covers h 0..15; group1 covers h 16..23 (rows 8..15 masked to zero).
__global__ void __launch_bounds__(256)
bbox_stage1(const float* __restrict__ X, const float* __restrict__ Pred,
            const float* __restrict__ Gt, float* __restrict__ ws) {
  const int b    = blockIdx.y;
  const int tid  = threadIdx.x;
  const int wave = tid >> 5;
  const int lane = tid & 31;
  const int hm   = lane & 15;

  const BoxParams bp0 = load_box(Pred, Gt, b, hm);
  const BoxParams bp1 = load_box(Pred, Gt, b, 16 + (hm & 7));  // clamped; masked below
  const float m1 = (hm < 8) ? 1.0f : 0.0f;

  v8f c_tp0 = {0,0,0,0,0,0,0,0}, c_pr0 = {0,0,0,0,0,0,0,0};
  v8f c_gt0 = {0,0,0,0,0,0,0,0}, c_ce0 = {0,0,0,0,0,0,0,0};
  v8f c_tp1 = {0,0,0,0,0,0,0,0}, c_pr1 = {0,0,0,0,0,0,0,0};
  v8f c_gt1 = {0,0,0,0,0,0,0,0}, c_ce1 = {0,0,0,0,0,0,0,0};

  const size_t base_pt = (size_t)b * N_;
  const int span = blockIdx.x * PTS_PER_BLOCK;

  for (int s = 0; s < STEPS; ++s) {
    const int n = span + (s * 8 + wave) * 4 + ((lane < 16) ? 0 : 2);
    const float* xa = X + (base_pt + n) * 9;
    const float* xb = xa + 9;
    __builtin_prefetch(xa + 9 * 32, 0, 1);  // next step's lines -> global_prefetch_b8
    const float a0 = xa[0], a1 = xa[1], a2 = xa[2];
    const float b0 = xb[0], b1 = xb[1], b2 = xb[2];

    float pA, gA, tA, cA, pB, gB, tB, cB;
    v2f av;

    eval_point(a0, a1, a2, bp0, 1.0f, pA, gA, tA, cA);
    eval_point(b0, b1, b2, bp0, 1.0f, pB, gB, tB, cB);
    av.x = tA; av.y = tB; c_tp0 = wmma_rowsum(av, c_tp0);
    av.x = pA; av.y = pB; c_pr0 = wmma_rowsum(av, c_pr0);
    av.x = gA; av.y = gB; c_gt0 = wmma_rowsum(av, c_gt0);
    av.x = cA; av.y = cB; c_ce0 = wmma_rowsum(av, c_ce0);

    eval_point(a0, a1, a2, bp1, m1, pA, gA, tA, cA);
    eval_point(b0, b1, b2, bp1, m1, pB, gB, tB, cB);
    av.x = tA; av.y = tB; c_tp1 = wmma_rowsum(av, c_tp1);
    av.x = pA; av.y = pB; c_pr1 = wmma_rowsum(av, c_pr1);
    av.x = gA; av.y = gB; c_gt1 = wmma_rowsum(av, c_gt1);
    av.x = cA; av.y = cB; c_ce1 = wmma_rowsum(av, c_ce1);
  }

  // Extract per-h row sums. C/D layout: VGPR r, lanes 0-15 -> M=r,
  // lanes 16-31 -> M=8+r; all columns identical (B = ones, C started at 0).
  // Each (wave,q,h) LDS slot has exactly one writer -> deterministic.
  __shared__ float lacc[8][4][H_];
#pragma unroll
  for (int r = 0; r < 8; ++r) {
    if (lane == 0) {
      lacc[wave][0][r] = c_tp0[r];
      lacc[wave][1][r] = c_pr0[r];
      lacc[wave][2][r] = c_gt0[r];
      lacc[wave][3][r] = c_ce0[r];
      lacc[wave][0][16 + r] = c_tp1[r];   // group1 rows 0..7 -> h = 16..23
      lacc[wave][1][16 + r] = c_pr1[r];
      lacc[wave][2][16 + r] = c_gt1[r];
      lacc[wave][3][16 + r] = c_ce1[r];
    }
    if (lane == 16) {
      lacc[wave][0][8 + r] = c_tp0[r];    // group0 rows 8..15
      lacc[wave][1][8 + r] = c_pr0[r];
      lacc[wave][2][8 + r] = c_gt0[r];
      lacc[wave][3][8 + r] = c_ce0[r];
    }
  }
  __syncthreads();
  if (tid < 4 * H_) {
    const int q = tid / H_, h = tid % H_;
    float sum = 0.0f;
#pragma unroll
    for (int w = 0; w < 8; ++w) sum += lacc[w][q][h];
    ws[(((size_t)b * H_ + h) * 4 + q) * NBLK + blockIdx.x] = sum;
  }
}

// Stage 2: 384 (b,h) scalars -> 4 outputs.
__global__ void __launch_bounds__(512)
bbox_finalize(const float* __restrict__ ws, const float* __restrict__ Pred,
              const float* __restrict__ Gt, float* __restrict__ out) {
  const int t = threadIdx.x;
  float vals[6] = {0, 0, 0, 0, 0, 0};  // helper, ce*h, iou*h, l2*h, 1-h, negd
  if (t < B_ * H_) {
    const int b = t / H_, h = t % H_;
    float tp = 0, sp = 0, sg = 0, ces = 0;
    const float* base = ws + (((size_t)b * H_ + h) * 4) * NBLK;
    for (int i = 0; i < NBLK; ++i) {
      tp  += base[0 * NBLK + i];
      sp  += base[1 * NBLK + i];
      sg  += base[2 * NBLK + i];
      ces += base[3 * NBLK + i];
    }
    const float* Yb = Gt   + (size_t)(b * H_ + h) * 6;
    const float* Pb = Pred + (size_t)(b * H_ + h) * 6;
    const float ysum = Yb[0] + Yb[1] + Yb[2] + Yb[3] + Yb[4] + Yb[5];
    const float helper = (ysum > 0.0f) ? 1.0f : 0.0f;
    const float iou = -(tp / (tp + (sp - tp) + (sg - tp) + 1e-6f));
    float l2 = 0.0f;
    for (int d = 0; d < 6; ++d) { float df = Yb[d] - Pb[d]; l2 += df * df; }
    l2 *= (1.0f / 6.0f);
    float dd = 0.0f;
    for (int d = 0; d < 3; ++d) { float df = Pb[d] - Pb[3 + d]; dd += df * df; }
    vals[0] = helper;
    vals[1] = ces * helper;
    vals[2] = iou * helper;
    vals[3] = l2 * helper;
    vals[4] = 1.0f - helper;
    vals[5] = (1.0f - helper) * dd;
  }
  __shared__ float red[512];
  float S[6];
  for (int q = 0; q < 6; ++q) {
    red[t] = vals[q];
    __syncthreads();
    for (int s = 256; s > 0; s >>= 1) {
      if (t < s) red[t] += red[t + s];
      __syncthreads();
    }
    S[q] = red[0];
    __syncthreads();
  }
  if (t == 0) {
    const float Sh = S[0];
    const float loss_ce  = S[1] / (Sh * (float)N_);
    const float loss_iou = S[2] / Sh;
    const float l2_neg   = S[5] / (S[4] + 1e-8f);
    const float loss_l2  = S[3] / Sh + l2_neg;
    out[0] = loss_ce + loss_l2 + loss_iou;  // LABEL == use_all_ce_l2_iou
    out[1] = loss_l2;
    out[2] = loss_ce;
    out[3] = loss_iou;
  }
}

extern "C" void kernel_launch(void* const* d_in, const int* in_sizes, int n_in,
                              void* d_out, int out_size, void* d_ws, size_t ws_size,
                              hipStream_t stream) {
  const float* X = (const float*)d_in[0];  // X_pc          (16, 40960, 9)
  const float* P = (const float*)d_in[1];  // y_bbvert_pred (16, 24, 2, 3)
  const float* G = (const float*)d_in[2];  // Y_bbvert      (16, 24, 2, 3)
  float* ws = (float*)d_ws;                // 16*24*4*40 floats = 240 KB, fully overwritten
  dim3 grid(NBLK, B_, 1);
  bbox_stage1<<<grid, 256, 0, stream>>>(X, P, G, ws);
  bbox_finalize<<<1, 512, 0, stream>>>(ws, P, G, (float*)d_out);
}